// Rroi_align_10771777978984
// MI455X (gfx1250) — compile-verified
//
#include <hip/hip_runtime.h>
#include <hip/hip_bf16.h>

#define NUMROI 32
#define PHI 32          // pooled_height (structural: defines out_size)
#define PWI 32          // pooled_width
#define WDIM 256        // features.shape[1]
#define HDIM 256        // features.shape[2]
#define PI_F 3.1415926535f

// One workgroup per ROI; 1024 threads = 32 wave32 waves; one bin per thread.
// Dynamic LDS holds the full 256x256 f32 slab features[0,0,:,:] (256 KB of the
// 320 KB WGP LDS). The async global->LDS copy is overlapped with the per-bin
// affine-transform math, which does not depend on the slab.
__global__ __launch_bounds__(1024) void rroi_align_kernel(
    const float* __restrict__ feat,
    const float* __restrict__ rois,
    const int*   __restrict__ ss_ptr,
    float*       __restrict__ out)
{
    extern __shared__ float slab[];   // 65536 floats, dynamic LDS base offset 0

    const int tid = (int)threadIdx.x;     // 0..1023
    const int roi = (int)blockIdx.x;      // 0..31

    // ---- Issue 256 KB slab -> LDS via CDNA5 async global->LDS b128 copies.
    // 1024 threads x 16 chunks x 16 B = 262144 B, fully coalesced. Do NOT wait
    // yet: the affine math below is independent of the slab and hides the copy.
    {
        const unsigned base = (unsigned)tid * 16u;
#pragma unroll
        for (int k = 0; k < 16; ++k) {
            const unsigned off = base + (unsigned)k * 16384u;
            // vdst = LDS byte address, vaddr = 32-bit offset, saddr = 64-bit base
            asm volatile("global_load_async_to_lds_b128 %0, %1, %2"
                         :
                         : "v"(off), "v"(off), "s"(feat)
                         : "memory");
        }
    }

    // ---- Per-ROI affine transform (reference: roi_idx = [0] + range(31)).
    const float* rp = rois + ((roi == 0) ? 0 : (roi - 1)) * 6;
    const float M1 = rp[1];
    const float M2 = rp[2];
    const float M3 = rp[3];
    const float M4 = rp[4];
    const float M5 = rp[5] * 180.0f * PI_F;
    const float ss = (float)(*ss_ptr);

    const float roi_pw = M4 / M3 * (float)PWI;
    const float dx = -roi_pw * 0.5f;
    const float dy = -((float)PHI) * 0.5f;
    const float Sx = M4 / roi_pw * ss;
    const float Sy = M3 / ((float)PHI * ss);
    const float Alpha = cosf(M5);
    const float Beta  = sinf(M5);
    const float Dx = M1 * ss;
    const float Dy = M2 * ss;
    const float M00 = Alpha * Sx;
    const float M01 = Beta * Sy;
    const float M02 = M00 * dx + M01 * dy + Dx;
    const float M10 = -Beta * Sx;
    const float M11 = Alpha * Sy;
    const float M12 = M10 * dx + M11 * dy + Dy;

    // Bin coordinates: out[roi, i, j]; x = j (fast), y = i (slow).
    const float x = (float)(tid & 31);
    const float y = (float)(tid >> 5);

    const float P0 = M00 * x         + M01 * y         + M02;
    const float P1 = M10 * x         + M11 * y         + M12;
    const float P2 = M00 * x         + M01 * (y + 1.f) + M02;
    const float P3 = M10 * x         + M11 * (y + 1.f) + M12;
    const float P4 = M00 * (x + 1.f) + M01 * y         + M02;
    const float P5 = M10 * (x + 1.f) + M11 * y         + M12;
    const float P6 = M00 * (x + 1.f) + M01 * (y + 1.f) + M02;
    const float P7 = M10 * (x + 1.f) + M11 * (y + 1.f) + M12;

    // jnp.round == round-half-even == rintf (RTNE default mode)
    const float leftMost   = fmaxf(rintf(fminf(fminf(P0, P2), fminf(P4, P6))), 0.0f);
    const float rightMost  = fminf(rintf(fmaxf(fmaxf(P0, P2), fmaxf(P4, P6))), (float)(WDIM - 1));
    const float topMost    = fmaxf(rintf(fminf(fminf(P1, P3), fminf(P5, P7))), 0.0f);
    const float bottomMost = fminf(rintf(fmaxf(fmaxf(P1, P3), fmaxf(P5, P7))), (float)(HDIM - 1));

    const float bcx = (leftMost + rightMost) * 0.5f;
    const float bcy = (topMost + bottomMost) * 0.5f;

    const float fl = floorf(bcx), fr = ceilf(bcx);
    const float ft = floorf(bcy), fb = ceilf(bcy);

    // JAX gather clamps OOB indices -> clamp to [0, 255]
    const int il = min(max((int)fl, 0), WDIM - 1);
    const int ir = min(max((int)fr, 0), WDIM - 1);
    const int it = min(max((int)ft, 0), HDIM - 1);
    const int ib = min(max((int)fb, 0), HDIM - 1);

    const float rx = bcx - fl;
    const float ry = bcy - ft;
    const float wlt = (1.0f - rx) * (1.0f - ry);
    const float wrt = rx * (1.0f - ry);
    const float wrb = rx * ry;
    const float wlb = (1.0f - rx) * ry;

    // ---- Now the slab is needed: drain the async copies and sync the WG.
    asm volatile("s_wait_asynccnt 0" ::: "memory");
    __syncthreads();

    // gv(a,b) = float(int32(row0[a,b])) — truncation toward zero.
    const float ltv = (float)(int)slab[il * HDIM + it];
    const float rtv = (float)(int)slab[ir * HDIM + it];
    const float lbv = (float)(int)slab[il * HDIM + ib];
    const float rbv = (float)(int)slab[ir * HDIM + ib];

    out[roi * (PHI * PWI) + tid] = ltv * wlt + rtv * wrt + rbv * wrb + lbv * wlb;
}

extern "C" void kernel_launch(void* const* d_in, const int* in_sizes, int n_in,
                              void* d_out, int out_size, void* d_ws, size_t ws_size,
                              hipStream_t stream) {
    // setup_inputs order: pooled_height, pooled_width, spatial_scale, features, rois
    const int*   ss   = (const int*)d_in[2];
    const float* feat = (const float*)d_in[3];
    const float* rois = (const float*)d_in[4];
    float*       out  = (float*)d_out;

    const size_t lds_bytes = (size_t)WDIM * HDIM * sizeof(float); // 256 KB
    rroi_align_kernel<<<dim3(NUMROI), dim3(PHI * PWI), lds_bytes, stream>>>(feat, rois, ss, out);
}